// GCNGuard_9363028706302
// MI455X (gfx1250) — compile-verified
//
#include <hip/hip_runtime.h>
#include <math.h>

// ---------------------------------------------------------------------------
// GCNGuard forward on gfx1250 (wave32). GEMMs use V_WMMA_F32_16X16X4_F32.
// Edge structure (rev-edge hash, CSR) built once; 3 layers share it.
// ---------------------------------------------------------------------------

typedef __attribute__((ext_vector_type(2))) float v2f;
typedef __attribute__((ext_vector_type(8))) float v8f;

#define HASH_BITS 21
#define HASH_SIZE (1u << HASH_BITS)
#define HASH_MASK (HASH_SIZE - 1u)
#define HASH_EMPTY 0xFFFFFFFFu

__device__ __forceinline__ float wave_sum(float v) {
#pragma unroll
  for (int o = 16; o > 0; o >>= 1) v += __shfl_xor(v, o, 32);
  return v;
}
__device__ __forceinline__ float wave_max(float v) {
#pragma unroll
  for (int o = 16; o > 0; o >>= 1) v = fmaxf(v, __shfl_xor(v, o, 32));
  return v;
}

// ---------------- structure build (once per launch) ----------------

__global__ void k_zero_i32(int* p, int n) {
  int i = blockIdx.x * blockDim.x + threadIdx.x;
  if (i < n) p[i] = 0;
}

__global__ void k_hash_init(unsigned* hkey, int n) {
  int i = blockIdx.x * blockDim.x + threadIdx.x;
  if (i < n) hkey[i] = HASH_EMPTY;
}

__global__ void k_hash_insert(const int* __restrict__ row, const int* __restrict__ col,
                              int E, unsigned N, unsigned* hkey, int* hval) {
  int e = blockIdx.x * blockDim.x + threadIdx.x;
  if (e >= E) return;
  unsigned k = (unsigned)row[e] * N + (unsigned)col[e];
  unsigned p = (k * 2654435761u) & HASH_MASK;
  for (;;) {
    unsigned prev = atomicCAS(&hkey[p], HASH_EMPTY, k);
    if (prev == HASH_EMPTY || prev == k) { hval[p] = e; break; }
    p = (p + 1u) & HASH_MASK;
  }
}

__global__ void k_revidx(const int* __restrict__ row, const int* __restrict__ col,
                         int E, unsigned N, const unsigned* __restrict__ hkey,
                         const int* __restrict__ hval, int* __restrict__ rev) {
  int e = blockIdx.x * blockDim.x + threadIdx.x;
  if (e >= E) return;
  unsigned k = (unsigned)col[e] * N + (unsigned)row[e];
  unsigned p = (k * 2654435761u) & HASH_MASK;
  int r = -1;
  for (;;) {
    unsigned cur = hkey[p];
    if (cur == k) { r = hval[p]; break; }
    if (cur == HASH_EMPTY) break;
    p = (p + 1u) & HASH_MASK;
  }
  rev[e] = r;
}

__global__ void k_count(const int* __restrict__ row, int E, int* cnt) {
  int e = blockIdx.x * blockDim.x + threadIdx.x;
  if (e < E) atomicAdd(&cnt[row[e]], 1);
}

// single-block exclusive scan (deterministic), n up to ~10^6
__global__ void k_scan(const int* __restrict__ cnt, int* __restrict__ off, int n) {
  __shared__ int temp[1024];
  __shared__ int running;
  int tid = threadIdx.x;
  if (tid == 0) running = 0;
  __syncthreads();
  for (int base = 0; base < n; base += 1024) {
    int v = (base + tid < n) ? cnt[base + tid] : 0;
    temp[tid] = v;
    __syncthreads();
    for (int o = 1; o < 1024; o <<= 1) {
      int t = (tid >= o) ? temp[tid - o] : 0;
      __syncthreads();
      temp[tid] += t;
      __syncthreads();
    }
    int incl = temp[tid];
    int r = running;
    if (base + tid < n) off[base + tid] = r + incl - v;
    int total = temp[1023];
    __syncthreads();
    if (tid == 0) running = r + total;
    __syncthreads();
  }
  if (tid == 0) off[n] = running;
}

__global__ void k_fill(const int* __restrict__ row, int E, const int* __restrict__ off,
                       int* cursor, int* __restrict__ csr) {
  int e = blockIdx.x * blockDim.x + threadIdx.x;
  if (e >= E) return;
  int r = row[e];
  int pos = atomicAdd(&cursor[r], 1);
  csr[off[r] + pos] = e;
}

// ---------------- per-layer attention ----------------

// one wave per node: row L2 norm; also zeroes rs/deg for this layer
__global__ void k_norms(const float* __restrict__ h, float* __restrict__ norms,
                        float* __restrict__ rs, float* __restrict__ deg, int n) {
  int wid = (blockIdx.x * blockDim.x + threadIdx.x) >> 5;
  int lane = threadIdx.x & 31;
  if (wid >= n) return;
  float4 v = ((const float4*)(h + (size_t)wid * 128))[lane];
  float ss = wave_sum(v.x * v.x + v.y * v.y + v.z * v.z + v.w * v.w);
  if (lane == 0) { norms[wid] = sqrtf(ss); rs[wid] = 0.f; deg[wid] = 0.f; }
}

// one wave per edge: cosine sim, threshold, accumulate row sums
__global__ void k_sim(const float* __restrict__ h, const float* __restrict__ norms,
                      const int* __restrict__ row, const int* __restrict__ col,
                      float* __restrict__ att, float* rs, int E) {
  int wid = (blockIdx.x * blockDim.x + threadIdx.x) >> 5;
  int lane = threadIdx.x & 31;
  if (wid >= E) return;
  int r = row[wid], c = col[wid];
  float4 a = ((const float4*)(h + (size_t)r * 128))[lane];
  float4 b = ((const float4*)(h + (size_t)c * 128))[lane];
  float d = wave_sum(a.x * b.x + a.y * b.y + a.z * b.z + a.w * b.w);
  if (lane == 0) {
    float s = d / (norms[r] * norms[c]);
    if (s < 0.1f) s = 0.f;
    att[wid] = s;
    if (s != 0.f) atomicAdd(&rs[r], s);
  }
}

__global__ void k_norm_att(float* __restrict__ att, const float* __restrict__ rs,
                           const int* __restrict__ row, int E) {
  int e = blockIdx.x * blockDim.x + threadIdx.x;
  if (e >= E) return;
  float r = rs[row[e]];
  att[e] = att[e] / (r > 0.f ? r : 1.f);
}

// learned binary drop + degree count + exp weights
__global__ void k_maskdeg(const float* __restrict__ att, const int* __restrict__ rev,
                          const int* __restrict__ row, float* __restrict__ w_e,
                          float* deg, const float* __restrict__ dW,
                          const float* __restrict__ dB, int E) {
  int e = blockIdx.x * blockDim.x + threadIdx.x;
  if (e >= E) return;
  float a = att[e];
  int ri = rev[e];
  float ar = (ri >= 0) ? att[ri] : 0.f;
  float z = a * dW[0] + ar * dW[1] + dB[0];     // sigmoid(z)>0.5 <=> z>0
  float am = (z > 0.f) ? a : 0.f;
  if (am != 0.f) {
    atomicAdd(&deg[row[e]], 1.f);
    w_e[e] = expf(am);
  } else {
    w_e[e] = 0.f;
  }
}

// ---------------- GEMM: S[n x 128] = A[n x 128] * W[128 x 128] ----------------
// one wave per 16x16 tile; 8 waves/block cover the 128 output columns.
// V_WMMA_F32_16X16X4_F32 register layouts per CDNA5 ISA 7.12.2:
//  A 16x4:  lanes 0-15 (M=lane) hold K={k,k+1} in v0,v1; lanes 16-31 hold K={k+2,k+3}
//  B 4x16:  v0: K=k (lanes 0-15) / K=k+2 (lanes 16-31); v1: K=k+1 / K=k+3; N=lane&15
//  C/D:     vj: M=j (lanes 0-15) / M=j+8 (lanes 16-31); N=lane&15
__global__ void k_gemm128(const float* __restrict__ A, const float* __restrict__ W,
                          float* __restrict__ S, int n) {
  int m0 = blockIdx.x * 16;
  if (m0 >= n) return;
  int wave = threadIdx.x >> 5;
  int lane = threadIdx.x & 31;
  int n0 = wave * 16;
  int lh = lane & 15;
  int kadd = (lane >> 4) << 1;  // 0 or 2
  const float* arow = A + (size_t)(m0 + lh) * 128;
  v8f c = {};
#pragma unroll 4
  for (int k0 = 0; k0 < 128; k0 += 4) {
    v2f a, b;
    a.x = arow[k0 + kadd];
    a.y = arow[k0 + kadd + 1];
    b.x = W[(k0 + kadd) * 128 + n0 + lh];
    b.y = W[(k0 + kadd + 1) * 128 + n0 + lh];
    c = __builtin_amdgcn_wmma_f32_16x16x4_f32(false, a, false, b, (short)0, c,
                                              false, false);
  }
  int mh = (lane >> 4) << 3;  // 0 or 8
#pragma unroll
  for (int j = 0; j < 8; j++)
    S[(size_t)(m0 + mh + j) * 128 + n0 + lh] = c[j];
}

// ---------------- fused aggregate + self-loop + bias + LN/ReLU or log_softmax ----
// one wave per node; mode 0: LayerNorm+ReLU, mode 1: log_softmax
__global__ void k_combine(const float* __restrict__ S, const float* __restrict__ w_e,
                          const float* __restrict__ deg, const int* __restrict__ off,
                          const int* __restrict__ csr, const int* __restrict__ col,
                          const float* __restrict__ bias, const float* __restrict__ g,
                          const float* __restrict__ bln, float* __restrict__ out,
                          int n, int mode) {
  int i = (blockIdx.x * blockDim.x + threadIdx.x) >> 5;
  int lane = threadIdx.x & 31;
  if (i >= n) return;
  float4 si = ((const float4*)(S + (size_t)i * 128))[lane];
  float4 acc = {0.f, 0.f, 0.f, 0.f};
  int p1 = off[i + 1];
  for (int p = off[i]; p < p1; ++p) {
    int e = csr[p];
    float w = w_e[e];
    if (w != 0.f) {
      float4 sc = ((const float4*)(S + (size_t)col[e] * 128))[lane];
      acc.x += w * sc.x; acc.y += w * sc.y; acc.z += w * sc.z; acc.w += w * sc.w;
    }
  }
  float wd = expf(1.f / (deg[i] + 1.f));
  float4 bi = ((const float4*)bias)[lane];
  float4 v;
  v.x = acc.x + wd * si.x + bi.x;
  v.y = acc.y + wd * si.y + bi.y;
  v.z = acc.z + wd * si.z + bi.z;
  v.w = acc.w + wd * si.w + bi.w;
  if (mode == 0) {
    float mu = wave_sum(v.x + v.y + v.z + v.w) * (1.f / 128.f);
    float dx = v.x - mu, dy = v.y - mu, dz = v.z - mu, dw = v.w - mu;
    float var = wave_sum(dx * dx + dy * dy + dz * dz + dw * dw) * (1.f / 128.f);
    float rstd = rsqrtf(var + 1e-5f);
    float4 gg = ((const float4*)g)[lane];
    float4 bb = ((const float4*)bln)[lane];
    v.x = fmaxf(dx * rstd * gg.x + bb.x, 0.f);
    v.y = fmaxf(dy * rstd * gg.y + bb.y, 0.f);
    v.z = fmaxf(dz * rstd * gg.z + bb.z, 0.f);
    v.w = fmaxf(dw * rstd * gg.w + bb.w, 0.f);
  } else {
    float m = wave_max(fmaxf(fmaxf(v.x, v.y), fmaxf(v.z, v.w)));
    float se = wave_sum(expf(v.x - m) + expf(v.y - m) + expf(v.z - m) + expf(v.w - m));
    float lse = logf(se) + m;
    v.x -= lse; v.y -= lse; v.z -= lse; v.w -= lse;
  }
  ((float4*)(out + (size_t)i * 128))[lane] = v;
}

// ---------------------------------------------------------------------------

extern "C" void kernel_launch(void* const* d_in, const int* in_sizes, int n_in,
                              void* d_out, int out_size, void* d_ws, size_t ws_size,
                              hipStream_t stream) {
  const float* x    = (const float*)d_in[0];
  const float* W0   = (const float*)d_in[1];
  const float* b0   = (const float*)d_in[2];
  const float* W1   = (const float*)d_in[3];
  const float* b1   = (const float*)d_in[4];
  const float* ln1g = (const float*)d_in[5];
  const float* ln1b = (const float*)d_in[6];
  const float* ln2g = (const float*)d_in[7];
  const float* ln2b = (const float*)d_in[8];
  const float* dW   = (const float*)d_in[9];
  const float* dB   = (const float*)d_in[10];
  const int*   row  = (const int*)d_in[11];
  const int*   col  = (const int*)d_in[12];

  const int N = in_sizes[0] / 128;
  const int E = in_sizes[11];

  // workspace carve (~82 MB)
  char* w = (char*)d_ws;
  auto carve = [&](size_t bytes) -> void* {
    void* p = (void*)w;
    w += (bytes + 255) & ~(size_t)255;
    return p;
  };
  float*    h      = (float*)carve((size_t)N * 128 * 4);
  float*    s      = (float*)carve((size_t)N * 128 * 4);
  float*    att    = (float*)carve((size_t)E * 4);
  float*    we     = (float*)carve((size_t)E * 4);
  float*    norms  = (float*)carve((size_t)N * 4);
  float*    rs     = (float*)carve((size_t)N * 4);
  float*    deg    = (float*)carve((size_t)N * 4);
  int*      rev    = (int*)carve((size_t)E * 4);
  int*      csr    = (int*)carve((size_t)E * 4);
  int*      off    = (int*)carve((size_t)(N + 1) * 4);
  int*      cursor = (int*)carve((size_t)N * 4);
  unsigned* hkey   = (unsigned*)carve((size_t)HASH_SIZE * 4);
  int*      hval   = (int*)carve((size_t)HASH_SIZE * 4);

  const int TB = 256;
  const int ebT = (E + TB - 1) / TB;       // per-edge-thread blocks
  const int ebW = (E + 7) / 8;             // per-edge-wave blocks (8 waves/block)
  const int nbW = (N + 7) / 8;             // per-node-wave blocks
  const int nbT = (N + TB - 1) / TB;

  // ---- build structure (constant across layers) ----
  k_hash_init<<<(HASH_SIZE + TB - 1) / TB, TB, 0, stream>>>(hkey, (int)HASH_SIZE);
  k_zero_i32<<<nbT, TB, 0, stream>>>(cursor, N);
  k_hash_insert<<<ebT, TB, 0, stream>>>(row, col, E, (unsigned)N, hkey, hval);
  k_revidx<<<ebT, TB, 0, stream>>>(row, col, E, (unsigned)N, hkey, hval, rev);
  k_count<<<ebT, TB, 0, stream>>>(row, E, cursor);
  k_scan<<<1, 1024, 0, stream>>>(cursor, off, N);
  k_zero_i32<<<nbT, TB, 0, stream>>>(cursor, N);
  k_fill<<<ebT, TB, 0, stream>>>(row, E, off, cursor, csr);

  // ---- one GNN layer ----
  auto layer = [&](const float* hin, const float* W, const float* bias,
                   const float* g, const float* bln, float* outp, int mode) {
    k_norms<<<nbW, TB, 0, stream>>>(hin, norms, rs, deg, N);
    k_sim<<<ebW, TB, 0, stream>>>(hin, norms, row, col, att, rs, E);
    k_norm_att<<<ebT, TB, 0, stream>>>(att, rs, row, E);
    k_maskdeg<<<ebT, TB, 0, stream>>>(att, rev, row, we, deg, dW, dB, E);
    k_gemm128<<<(N + 15) / 16, TB, 0, stream>>>(hin, W, s, N);
    k_combine<<<nbW, TB, 0, stream>>>(s, we, deg, off, csr, col, bias, g, bln,
                                      outp, N, mode);
  };

  layer(x, W0, b0, ln1g, ln1b, h, 0);                   // layer 0: LN1 + ReLU
  layer(h, W1, b1, ln2g, ln2b, h, 0);                   // layer 1: LN2 + ReLU
  layer(h, W1, b1, ln2g, ln2b, (float*)d_out, 1);       // final (reuses W1,b1): log_softmax
}